// TripleBranchWithSpecificConv_bak_67405216744034
// MI455X (gfx1250) — compile-verified
//
#include <hip/hip_runtime.h>

// Fused dynamic-kernel conv:
// out[b,oc,p] = sigmoid(att[b,p]) * (sum_k exp(w[b,oc,k,p]) * patch[b,k,p]) / (sum_k exp(w[b,oc,k,p]))
// Memory-bound on moire_raw (629 MB, read once, NT-hinted). WMMA f32 16x16x4 with
// A=ones performs the exact 4-slot cross-lane reduction of the softmax numerator
// and denominator in one matrix op each.

typedef __attribute__((ext_vector_type(2))) float v2f;
typedef __attribute__((ext_vector_type(8))) float v8f;

#define BATCH 4
#define CIN   3
#define HH    512
#define WW    512
#define MHh   128
#define MWw   128
#define KKDIM 75    // C*K*K = 3*25
#define OCH   32
#define PLANE (MHh * MWw)   // 16384

__global__ __launch_bounds__(256) void moire_fused_kernel(
    const float* __restrict__ x,       // [4,3,512,512]
    const float* __restrict__ moire,   // [4,2400,128,128]
    const float* __restrict__ att,     // [4,1,128,128]
    float* __restrict__ out)           // [4,32,128,128]
{
    __shared__ float p_lds[16 * 76];   // 16 pixels x 75 patch vals (stride 76: bank-conflict free)
    __shared__ float att_lds[16];

    const int tid = threadIdx.x;
    const int bx  = blockIdx.x;        // 4096 blocks = 4 * 128 * 8
    const int mwb = bx & 7;            // mw tile (16 pixels each)
    const int mh  = (bx >> 3) & 127;
    const int b   = bx >> 10;
    const int pixBase = mwb * 16;

    // ---- Phase 1: gather 16x75 unfold patches (k = c*25 + kh*5 + kw, pad=2, stride=4) ----
    for (int idx = tid; idx < 16 * KKDIM; idx += 256) {
        int pix = idx / KKDIM;
        int k   = idx - pix * KKDIM;
        int c   = k / 25;
        int rem = k - c * 25;
        int kh  = rem / 5;
        int kw  = rem - kh * 5;
        int ih  = mh * 4 - 2 + kh;
        int iw  = (pixBase + pix) * 4 - 2 + kw;
        float v = 0.0f;
        if ((unsigned)ih < (unsigned)HH && (unsigned)iw < (unsigned)WW) {
            v = x[(((long)b * CIN + c) * HH + ih) * (long)WW + iw];
        }
        p_lds[pix * 76 + k] = v;
    }
    if (tid < 16) {
        float a = att[(long)b * PLANE + mh * MWw + pixBase + tid];
        // sigmoid via exp2 (exact: exp(x) = exp2(x*log2e))
        att_lds[tid] = 1.0f / (1.0f + __builtin_amdgcn_exp2f(-a * 1.44269504088896340736f));
    }
    __syncthreads();

    // ---- Phase 2: streaming softmax-weighted reduction ----
    const int lane = tid & 31;
    const int wave = tid >> 5;         // 8 waves
    const int n    = lane & 15;        // pixel within tile
    const int half = lane >> 4;        // selects k-segment pair
    const int pixIdx = mh * MWw + pixBase + n;
    const float L2E = 1.44269504088896340736f;

    // k-segments: slot s in [0,4) covers k = s*19 .. min(s*19+18, 74); slot = 2*half + vgpr
    const int kA0 = (2 * half) * 19;       // 0 or 38
    const int kB0 = (2 * half + 1) * 19;   // 19 or 57 (57..75 -> last element masked)

    const v2f ones = {1.0f, 1.0f};

    for (int i = 0; i < 4; ++i) {
        const int oc = wave + 8 * i;       // 8 waves x 4 -> all 32 out-channels
        const float* wbase = moire + ((long)b * OCH + oc) * (long)KKDIM * PLANE + pixIdx;

        float sA = 0.0f, sB = 0.0f, dA = 0.0f, dB = 0.0f;
        for (int t = 0; t < 19; ++t) {
            int ka  = kA0 + t;                         // always < 75
            int kb  = kB0 + t;                         // half==1, t==18 -> 75 (invalid)
            int kbc = (kb < KKDIM) ? kb : (KKDIM - 1); // clamp address, mask value
            // streaming loads: don't pollute L2 (moire is 629MB, single-use)
            float w0 = __builtin_nontemporal_load(wbase + (long)ka  * PLANE);
            float w1 = __builtin_nontemporal_load(wbase + (long)kbc * PLANE);
            float e0 = __builtin_amdgcn_exp2f(w0 * L2E);
            float e1 = (kb < KKDIM) ? __builtin_amdgcn_exp2f(w1 * L2E) : 0.0f;
            float p0 = p_lds[n * 76 + ka];
            float p1 = p_lds[n * 76 + kbc];
            sA += e0;  sB += e1;
            dA = fmaf(e0, p0, dA);
            dB = fmaf(e1, p1, dB);
        }

        // WMMA reduction: A = ones => C[m,n] = sum of the 4 B-slots of column n
        // (lane n v0/v1 + lane n+16 v0/v1) = full sum over k = 0..74. Order of
        // slot->k mapping is irrelevant since all slots are summed.
        v2f bs = {sA, sB};
        v2f bd = {dA, dB};
        v8f accS = {};
        v8f accD = {};
        accS = __builtin_amdgcn_wmma_f32_16x16x4_f32(false, ones, false, bs,
                                                     (short)0, accS, false, false);
        accD = __builtin_amdgcn_wmma_f32_16x16x4_f32(false, ones, false, bd,
                                                     (short)0, accD, false, false);
        float ssum = accS[0];   // column n sum (all rows equal); lane holds N = lane&15
        float dsum = accD[0];
        float val  = att_lds[n] * dsum / ssum;

        if (half == 0) {        // lanes 0..15: one coalesced 64B store per oc
            out[((long)b * OCH + oc) * PLANE + pixIdx] = val;
        }
    }
}

extern "C" void kernel_launch(void* const* d_in, const int* in_sizes, int n_in,
                              void* d_out, int out_size, void* d_ws, size_t ws_size,
                              hipStream_t stream) {
    const float* x     = (const float*)d_in[0];   // [4,3,512,512]
    const float* moire = (const float*)d_in[1];   // [4,2400,128,128]
    const float* att   = (const float*)d_in[2];   // [4,1,128,128]
    float* out = (float*)d_out;                   // [4,32,128,128]
    (void)in_sizes; (void)n_in; (void)out_size; (void)d_ws; (void)ws_size;

    dim3 grid(BATCH * MHh * (MWw / 16));          // 4096 blocks
    dim3 block(256);                              // 8 wave32
    hipLaunchKernelGGL(moire_fused_kernel, grid, block, 0, stream, x, moire, att, out);
}